// ChooseDestAndUpdate_14628658610881
// MI455X (gfx1250) — compile-verified
//
#include <hip/hip_runtime.h>
#include <hip/hip_bf16.h>
#include <math.h>

#define HDIM 128

typedef __attribute__((ext_vector_type(2))) float v2f;
typedef __attribute__((ext_vector_type(8))) float v8f;

// D = A(16x4) * B(4x16) + C, fp32 WMMA (CDNA5 v_wmma_f32_16x16x4_f32)
__device__ __forceinline__ v8f wmma4(v2f a, v2f b, v8f c) {
  return __builtin_amdgcn_wmma_f32_16x16x4_f32(
      /*neg_a=*/false, a, /*neg_b=*/false, b,
      /*c_mod=*/(short)0, c, /*reuse_a=*/false, /*reuse_b=*/false);
}

// ---------------------------------------------------------------------------
// Pack a (K x Ncols) B-matrix into WMMA fragment order:
//   dst[((k0/4)*NT + nt)*32 + lane] = { B[k0+kh][nt*16+n], B[k0+kh+1][nt*16+n] }
// with n = lane&15, kh = (lane<16)?0:2.  transpose=1 means B[k][j]=src[j*ld+k].
// After packing, each wave's B fragment is ONE contiguous global_load_b64.
// ---------------------------------------------------------------------------
__global__ void pack_kernel(const float* __restrict__ src, v2f* __restrict__ dst,
                            int K, int NT, int ldsrc, int transpose) {
  int id = blockIdx.x * blockDim.x + threadIdx.x;
  int total = (K >> 2) * NT * 32;
  if (id >= total) return;
  int lane  = id & 31;
  int nt    = (id >> 5) % NT;
  int kstep = id / (32 * NT);
  int k     = kstep * 4 + ((lane < 16) ? 0 : 2);
  int col   = nt * 16 + (lane & 15);
  v2f v;
  if (transpose) {
    v.x = src[(long)col * ldsrc + k];
    v.y = src[(long)col * ldsrc + k + 1];
  } else {
    v.x = src[(long)k * ldsrc + col];
    v.y = src[(long)(k + 1) * ldsrc + col];
  }
  dst[id] = v;
}

// ---------------------------------------------------------------------------
// prep: int32 row/col (+2 appended edges), extended edge_attr, output copies.
// ---------------------------------------------------------------------------
__global__ void prep_kernel(const long long* __restrict__ ei,
                            const float* __restrict__ ea,
                            const int* __restrict__ destp,
                            int* __restrict__ rowi, int* __restrict__ coli,
                            float* __restrict__ eaext,
                            float* __restrict__ out_ei, float* __restrict__ out_ea,
                            int E, int N) {
  int i = blockIdx.x * blockDim.x + threadIdx.x;
  int Etot = E + 2;
  if (i >= Etot) return;
  long long r, c; float a;
  if (i < E)        { r = ei[i];     c = ei[(long)E + i]; a = ea[i]; }
  else if (i == E)  { r = N - 1;     c = destp[0];        a = 1.0f;  }
  else              { r = destp[0];  c = N - 1;           a = 1.0f;  }
  rowi[i] = (int)r; coli[i] = (int)c; eaext[i] = a;
  out_ei[i] = (float)r; out_ei[Etot + i] = (float)c; out_ea[i] = a;
}

__global__ void zero_kernel(float* __restrict__ p, long n) {
  long i = (long)blockIdx.x * blockDim.x + threadIdx.x;
  long stride = (long)gridDim.x * blockDim.x;
  for (; i < n; i += stride) p[i] = 0.0f;
}

// ---------------------------------------------------------------------------
// scores: one wave (32 lanes) per destination node.
// ---------------------------------------------------------------------------
__global__ __launch_bounds__(256) void scores_kernel(
    const float* __restrict__ x, const float* __restrict__ wcd,
    const float* __restrict__ bcd, float* __restrict__ scores, int M, int N) {
  int gwave = (blockIdx.x * blockDim.x + threadIdx.x) >> 5;
  int lane = threadIdx.x & 31;
  if (gwave >= M) return;
  const float* xr = x + (long)gwave * HDIM;
  const float* xs = x + (long)(N - 1) * HDIM;
  float s = 0.0f;
  for (int k = lane; k < HDIM; k += 32)
    s += xr[k] * wcd[k] + xs[k] * wcd[HDIM + k];
  for (int off = 16; off > 0; off >>= 1) s += __shfl_down(s, off, 32);
  if (lane == 0) scores[gwave] = s + bcd[0];
}

__global__ __launch_bounds__(256) void lse_kernel(
    const float* __restrict__ scores, int M, const int* __restrict__ destp,
    float* __restrict__ out_lp) {
  __shared__ float red[256];
  int tid = threadIdx.x;
  float mx = -3.402823466e38f;
  for (int i = tid; i < M; i += 256) mx = fmaxf(mx, scores[i]);
  red[tid] = mx; __syncthreads();
  for (int s = 128; s > 0; s >>= 1) {
    if (tid < s) red[tid] = fmaxf(red[tid], red[tid + s]);
    __syncthreads();
  }
  mx = red[0]; __syncthreads();
  float sum = 0.0f;
  for (int i = tid; i < M; i += 256) sum += __expf(scores[i] - mx);
  red[tid] = sum; __syncthreads();
  for (int s = 128; s > 0; s >>= 1) {
    if (tid < s) red[tid] += red[tid + s];
    __syncthreads();
  }
  if (tid == 0) out_lp[0] = scores[destp[0]] - mx - __logf(red[0]);
}

// ---------------------------------------------------------------------------
// messages + scatter: one block per 32-edge tile (2 M-subtiles per wave so
// every packed B fragment is reused twice). 8 waves x 16 output cols.
// messages[e] = x[col[e]]@W[0:128] + x[row[e]]@W[128:256] + e_attr[e]*W[256]+b
// ---------------------------------------------------------------------------
__global__ __launch_bounds__(256) void msg_kernel(
    const float* __restrict__ x, const int* __restrict__ rowi,
    const int* __restrict__ coli, const float* __restrict__ eaext,
    const v2f* __restrict__ Wp /* packed (K=256, NT=8) */,
    const float* __restrict__ W /* raw (257,128), for row 256 */,
    const float* __restrict__ bias,
    float* __restrict__ agg, int nEdges) {
  __shared__ float A0[32][HDIM];   // x[col]
  __shared__ float A1[32][HDIM];   // x[row]
  __shared__ int   eRow[32];
  __shared__ int   eCol[32];
  __shared__ float eAtt[32];

  const int base = blockIdx.x * 32;
  const int tid  = threadIdx.x;

  if (tid < 32) {
    int e = base + tid;
    bool ok = (e < nEdges);
    eRow[tid] = ok ? rowi[e] : -1;
    eCol[tid] = ok ? coli[e] : -1;
    eAtt[tid] = ok ? eaext[e] : 0.0f;
  }
  __syncthreads();

  for (int i = tid; i < 2 * 32 * HDIM; i += 256) {
    int mat = i >> 12;            // 0: A0, 1: A1
    int er  = (i >> 7) & 31;
    int k   = i & (HDIM - 1);
    int src = mat ? eRow[er] : eCol[er];
    float v = (src >= 0) ? x[(long)src * HDIM + k] : 0.0f;
    if (mat) A1[er][k] = v; else A0[er][k] = v;
  }
  __syncthreads();

  const int lane  = tid & 31;
  const int wv    = tid >> 5;              // wave id = output col tile (NT=8)
  const int nb    = wv * 16;
  const int n     = lane & 15;
  const int khalf = (lane < 16) ? 0 : 2;
  const int mr    = lane & 15;

  v8f acc0 = {}, acc1 = {};
  // K-steps 0..31: A0 vs W rows [0,128); K-steps 32..63: A1 vs W rows [128,256)
  for (int ks = 0; ks < 32; ++ks) {
    int k = ks * 4 + khalf;
    v2f b = Wp[((long)ks * 8 + wv) * 32 + lane];
    v2f a0, a1;
    a0.x = A0[mr][k];      a0.y = A0[mr][k + 1];
    a1.x = A0[16 + mr][k]; a1.y = A0[16 + mr][k + 1];
    acc0 = wmma4(a0, b, acc0);
    acc1 = wmma4(a1, b, acc1);
  }
  for (int ks = 32; ks < 64; ++ks) {
    int k = (ks - 32) * 4 + khalf;
    v2f b = Wp[((long)ks * 8 + wv) * 32 + lane];
    v2f a0, a1;
    a0.x = A1[mr][k];      a0.y = A1[mr][k + 1];
    a1.x = A1[16 + mr][k]; a1.y = A1[16 + mr][k + 1];
    acc0 = wmma4(a0, b, acc0);
    acc1 = wmma4(a1, b, acc1);
  }

  const float wlast = W[(long)256 * HDIM + nb + n];  // edge_attr coefficient
  const float bcol  = bias[nb + n];
  #pragma unroll
  for (int j = 0; j < 8; ++j) {
    int m0 = j + ((lane < 16) ? 0 : 8);
    int e0 = base + m0;
    if (e0 < nEdges)
      atomicAdd(&agg[(long)eRow[m0] * HDIM + nb + n],
                acc0[j] + eAtt[m0] * wlast + bcol);
    int m1 = 16 + m0;
    int e1 = base + m1;
    if (e1 < nEdges)
      atomicAdd(&agg[(long)eRow[m1] * HDIM + nb + n],
                acc1[j] + eAtt[m1] * wlast + bcol);
  }
}

// ---------------------------------------------------------------------------
// GRU: one block per 32-node tile (2 M-subtiles). inp = [agg | h] (K=256).
// gi = inp @ Wih.T ; gh = h @ Whh.T ; r/z/n gates ; x' = (1-z)*n + z*h
// Wave w owns gate columns [16w,16w+16) of r, z and n -> packed col tiles
// w, w+8, w+16 of the (Ncols=384) transposed weights. 12 accumulators.
// ---------------------------------------------------------------------------
__global__ __launch_bounds__(256) void gru_kernel(
    const float* __restrict__ agg, const float* __restrict__ xin,
    const v2f* __restrict__ WihP /* packed K=256, NT=24 */,
    const v2f* __restrict__ WhhP /* packed K=128, NT=24 */,
    const float* __restrict__ bih, const float* __restrict__ bhh,
    float* __restrict__ xout, int N) {
  __shared__ float A[32][256];   // [agg(128) | h(128)] per node row

  const int base = blockIdx.x * 32;
  const int tid  = threadIdx.x;

  for (int i = tid; i < 32 * 256; i += 256) {
    int r = i >> 8;
    int k = i & 255;
    int node = base + r;
    float v = 0.0f;
    if (node < N)
      v = (k < HDIM) ? agg[(long)node * HDIM + k]
                     : xin[(long)node * HDIM + (k - HDIM)];
    A[r][k] = v;
  }
  __syncthreads();

  const int lane  = tid & 31;
  const int wv    = tid >> 5;
  const int nb    = wv * 16;               // gate column base (0..112)
  const int n     = lane & 15;
  const int khalf = (lane < 16) ? 0 : 2;
  const int mr    = lane & 15;

  v8f gr0 = {}, gz0 = {}, gn0 = {}, hr0 = {}, hz0 = {}, hn0 = {};
  v8f gr1 = {}, gz1 = {}, gn1 = {}, hr1 = {}, hz1 = {}, hn1 = {};

  // gi = inp @ Wih.T  (K = 256, packed NT = 24; tiles wv, wv+8, wv+16)
  for (int ks = 0; ks < 64; ++ks) {
    int k = ks * 4 + khalf;
    long rowoff = (long)ks * 24 * 32 + lane;
    v2f br = WihP[rowoff + (wv)      * 32];
    v2f bz = WihP[rowoff + (wv + 8)  * 32];
    v2f bn = WihP[rowoff + (wv + 16) * 32];
    v2f a0, a1;
    a0.x = A[mr][k];       a0.y = A[mr][k + 1];
    a1.x = A[16 + mr][k];  a1.y = A[16 + mr][k + 1];
    gr0 = wmma4(a0, br, gr0);  gr1 = wmma4(a1, br, gr1);
    gz0 = wmma4(a0, bz, gz0);  gz1 = wmma4(a1, bz, gz1);
    gn0 = wmma4(a0, bn, gn0);  gn1 = wmma4(a1, bn, gn1);
  }

  // gh = h @ Whh.T  (K = 128, packed NT = 24)
  for (int ks = 0; ks < 32; ++ks) {
    int k = ks * 4 + khalf;
    long rowoff = (long)ks * 24 * 32 + lane;
    v2f br = WhhP[rowoff + (wv)      * 32];
    v2f bz = WhhP[rowoff + (wv + 8)  * 32];
    v2f bn = WhhP[rowoff + (wv + 16) * 32];
    v2f a0, a1;
    a0.x = A[mr][HDIM + k];      a0.y = A[mr][HDIM + k + 1];
    a1.x = A[16 + mr][HDIM + k]; a1.y = A[16 + mr][HDIM + k + 1];
    hr0 = wmma4(a0, br, hr0);  hr1 = wmma4(a1, br, hr1);
    hz0 = wmma4(a0, bz, hz0);  hz1 = wmma4(a1, bz, hz1);
    hn0 = wmma4(a0, bn, hn0);  hn1 = wmma4(a1, bn, hn1);
  }

  const int col = nb + n;
  const float b_ir = bih[col], b_iz = bih[HDIM + col], b_in = bih[256 + col];
  const float b_hr = bhh[col], b_hz = bhh[HDIM + col], b_hn = bhh[256 + col];

  #pragma unroll
  for (int j = 0; j < 8; ++j) {
    int m0 = j + ((lane < 16) ? 0 : 8);
    {
      int node = base + m0;
      if (node < N) {
        float r  = 1.0f / (1.0f + __expf(-(gr0[j] + b_ir + hr0[j] + b_hr)));
        float z  = 1.0f / (1.0f + __expf(-(gz0[j] + b_iz + hz0[j] + b_hz)));
        float nn = tanhf(gn0[j] + b_in + r * (hn0[j] + b_hn));
        float h0 = A[m0][HDIM + col];
        xout[(long)node * HDIM + col] = (1.0f - z) * nn + z * h0;
      }
    }
    {
      int m1 = 16 + m0;
      int node = base + m1;
      if (node < N) {
        float r  = 1.0f / (1.0f + __expf(-(gr1[j] + b_ir + hr1[j] + b_hr)));
        float z  = 1.0f / (1.0f + __expf(-(gz1[j] + b_iz + hz1[j] + b_hz)));
        float nn = tanhf(gn1[j] + b_in + r * (hn1[j] + b_hn));
        float h0 = A[m1][HDIM + col];
        xout[(long)node * HDIM + col] = (1.0f - z) * nn + z * h0;
      }
    }
  }
}

// ---------------------------------------------------------------------------
extern "C" void kernel_launch(void* const* d_in, const int* in_sizes, int n_in,
                              void* d_out, int out_size, void* d_ws, size_t ws_size,
                              hipStream_t stream) {
  const float*     x     = (const float*)d_in[0];
  const long long* ei    = (const long long*)d_in[1];
  const float*     ea    = (const float*)d_in[2];
  const int*       destp = (const int*)d_in[3];
  const float*     wcd   = (const float*)d_in[4];
  const float*     bcd   = (const float*)d_in[5];
  const float*     msgW  = (const float*)d_in[6];   // (2, 257, 128)
  const float*     msgb  = (const float*)d_in[7];   // (2, 128)
  const float*     Wih   = (const float*)d_in[8];   // (2, 384, 256)
  const float*     Whh   = (const float*)d_in[9];   // (2, 384, 128)
  const float*     bih   = (const float*)d_in[10];  // (2, 384)
  const float*     bhh   = (const float*)d_in[11];  // (2, 384)

  const int N    = in_sizes[0] / HDIM;
  const int E    = in_sizes[2];
  const int Etot = E + 2;
  const int M    = N - 1;

  // workspace carve-up (256B aligned)
  char* ws = (char*)d_ws;
  size_t cur = 0;
  auto alloc = [&](size_t bytes) {
    size_t off = cur;
    cur += (bytes + 255) & ~size_t(255);
    return (void*)(ws + off);
  };
  float* scores = (float*)alloc(sizeof(float) * (size_t)N);
  int*   rowi   = (int*)  alloc(sizeof(int)   * (size_t)Etot);
  int*   coli   = (int*)  alloc(sizeof(int)   * (size_t)Etot);
  float* eaext  = (float*)alloc(sizeof(float) * (size_t)Etot);
  float* aggbuf = (float*)alloc(sizeof(float) * (size_t)N * HDIM);
  // packed weights: msgW (K=256,NT=8), Wih.T (K=256,NT=24), Whh.T (K=128,NT=24)
  const int nMsgP = 64 * 8  * 32;
  const int nIhP  = 64 * 24 * 32;
  const int nHhP  = 32 * 24 * 32;
  v2f* msgWp[2], *WihP[2], *WhhP[2];
  for (int t = 0; t < 2; ++t) {
    msgWp[t] = (v2f*)alloc(sizeof(v2f) * (size_t)nMsgP);
    WihP[t]  = (v2f*)alloc(sizeof(v2f) * (size_t)nIhP);
    WhhP[t]  = (v2f*)alloc(sizeof(v2f) * (size_t)nHhP);
  }
  (void)ws_size;

  // output layout (flat float32): x | edge_index | edge_attr | log_prob
  float* out_x  = (float*)d_out;
  float* out_ei = out_x + (size_t)N * HDIM;
  float* out_ea = out_ei + (size_t)2 * Etot;
  float* out_lp = out_ea + (size_t)Etot;
  (void)out_size;

  // 1) edges / attrs / int32 index arrays + weight packing
  prep_kernel<<<(Etot + 255) / 256, 256, 0, stream>>>(
      ei, ea, destp, rowi, coli, eaext, out_ei, out_ea, E, N);
  for (int t = 0; t < 2; ++t) {
    pack_kernel<<<(nMsgP + 255) / 256, 256, 0, stream>>>(
        msgW + (size_t)t * 257 * HDIM, msgWp[t], 256, 8, HDIM, 0);
    pack_kernel<<<(nIhP + 255) / 256, 256, 0, stream>>>(
        Wih + (size_t)t * 384 * 256, WihP[t], 256, 24, 256, 1);
    pack_kernel<<<(nHhP + 255) / 256, 256, 0, stream>>>(
        Whh + (size_t)t * 384 * HDIM, WhhP[t], 128, 24, HDIM, 1);
  }

  // 2) log_prob
  scores_kernel<<<(M + 7) / 8, 256, 0, stream>>>(x, wcd, bcd, scores, M, N);
  lse_kernel<<<1, 256, 0, stream>>>(scores, M, destp, out_lp);

  // 3) two message-passing + GRU rounds
  const int msgBlocks = (Etot + 31) / 32;
  const int gruBlocks = (N + 31) / 32;
  const long aggN = (long)N * HDIM;

  for (int t = 0; t < 2; ++t) {
    const float* Wt   = msgW + (size_t)t * 257 * HDIM;
    const float* bt   = msgb + (size_t)t * HDIM;
    const float* biht = bih + (size_t)t * 384;
    const float* bhht = bhh + (size_t)t * 384;
    const float* xin  = (t == 0) ? x : out_x;  // round 1 updates x in place

    zero_kernel<<<2048, 256, 0, stream>>>(aggbuf, aggN);
    msg_kernel<<<msgBlocks, 256, 0, stream>>>(
        xin, rowi, coli, eaext, msgWp[t], Wt, bt, aggbuf, Etot);
    gru_kernel<<<gruBlocks, 256, 0, stream>>>(
        aggbuf, xin, WihP[t], WhhP[t], biht, bhht, out_x, N);
  }
}